// HierachicalRNN_89069031784483
// MI455X (gfx1250) — compile-verified
//
#include <hip/hip_runtime.h>
#include <hip/hip_bf16.h>
#include <math.h>

// ---------------- CDNA5 / gfx1250 notes ----------------
// wave32 only; WMMA f32 16x16x4 used for all K=16 input-projection GEMMs.
// EXEC must be all-1s at WMMA: guards are wave-uniform; output buffers are
// padded to whole 16-row tiles so the store epilogue is branch-free.

typedef float v8f __attribute__((ext_vector_type(8)));
typedef float v2f __attribute__((ext_vector_type(2)));

#define VOCAB 136167
#define VPAD  136176   // ceil(VOCAB/16)*16
// B=32 F=2 HK=16 L=32 W=32 E=16 H=8 D=16, 3H=24

__device__ __forceinline__ float sigmoidf_(float x) {
    return 1.0f / (1.0f + __expf(-x));
}

// ---------------------------------------------------------------------------
// Fused attention matrices: Mc = cW @ aW (16x16), cc = cW @ ab (16)
// s[t,g,b,d] = sum_k ow[t,g,b,k]*Mc[d,k] + cc[d]
// ---------------------------------------------------------------------------
__global__ __launch_bounds__(256) void attn_prep(
    const float* __restrict__ aW, const float* __restrict__ ab,
    const float* __restrict__ cW, float* __restrict__ Mc, float* __restrict__ cc)
{
    int tid = threadIdx.x;          // 256
    int i = tid >> 4, j = tid & 15;
    float s = 0.f;
    #pragma unroll
    for (int m = 0; m < 16; ++m) s += cW[i * 16 + m] * aW[m * 16 + j];
    Mc[tid] = s;
    if (j == 0) {
        float t = 0.f;
        #pragma unroll
        for (int m = 0; m < 16; ++m) t += cW[i * 16 + m] * ab[m];
        cc[i] = t;
    }
}

// ---------------------------------------------------------------------------
// WMMA GEMM: out[Mpad,48] = A[Mrows,16] @ Wih[48,16]^T + bih[48]
// One wave per 16x16 output tile; K=16 via 4 x V_WMMA_F32_16X16X4_F32.
//   A frag  (16x4 f32): lanes 0-15 row M, v0=K0 v1=K1; lanes 16-31 v0=K2 v1=K3
//   B frag  (4x16 f32): lanes 0-15 col N, v0=K0 v1=K1; lanes 16-31 v0=K2 v1=K3
//   C/D: vgpr i -> (M = i + 8*(lane>=16), N = lane&15)
// out must have ceil(Mrows/16)*16 rows (padded rows get clamped-row garbage,
// never read downstream) -> branch-free store epilogue, no EXEC churn.
// ---------------------------------------------------------------------------
__global__ __launch_bounds__(256) void gemm_gi(
    const float* __restrict__ A, int Mrows,
    const float* __restrict__ Wih,   // [48][16] flat ((dir*24+j)*16+k)
    const float* __restrict__ bih,   // [48]
    float* __restrict__ out)         // [ceil(Mrows/16)*16][48]
{
    int wave = (blockIdx.x * blockDim.x + threadIdx.x) >> 5;  // global wave id
    int lane = threadIdx.x & 31;
    int mtiles = (Mrows + 15) >> 4;
    if (wave >= mtiles * 3) return;          // wave-uniform guard (EXEC stays all-1)
    int mtile = wave / 3, ntile = wave % 3;

    int hi = lane >> 4;                      // half-wave select
    int lr = lane & 15;
    int arow = mtile * 16 + lr;
    int arow_c = (arow < Mrows) ? arow : (Mrows - 1);  // clamp OOB loads only
    int ncol = ntile * 16 + lr;              // < 48

    v8f acc;
    float bias = bih[ncol];
    #pragma unroll
    for (int i = 0; i < 8; ++i) acc[i] = bias;

    const float* Ar = A + (size_t)arow_c * 16;
    const float* Wr = Wih + ncol * 16;
    #pragma unroll
    for (int kc = 0; kc < 4; ++kc) {
        int k0 = kc * 4 + hi * 2;
        v2f a, b;
        a.x = Ar[k0];
        a.y = Ar[k0 + 1];
        b.x = Wr[k0];
        b.y = Wr[k0 + 1];
        acc = __builtin_amdgcn_wmma_f32_16x16x4_f32(
                  false, a, false, b, (short)0, acc, false, false);
    }

    // rows mtile*16 + hi*8 + i, col ncol; single base + immediate offsets
    float* o = out + (size_t)(mtile * 16 + hi * 8) * 48 + ncol;
    #pragma unroll
    for (int i = 0; i < 8; ++i) o[i * 48] = acc[i];
}

// ---------------------------------------------------------------------------
// Word-level bidirectional GRU. 65536 threads = (dir, g=1024 groups, b=32).
// gi comes from the vocab-precomputed embproj[token][48] (L2-resident).
// ow layout: [t][g][b][16], fwd in d 0..7, bwd in d 8..15.
// ---------------------------------------------------------------------------
__global__ __launch_bounds__(256) void gru_word(
    const int* __restrict__ idx,           // [B,F,HK,L,W]
    const float* __restrict__ embproj,     // [VPAD][48]
    const float* __restrict__ Whh,         // [2][24][8]
    const float* __restrict__ bhh,         // [2][24]
    const float* __restrict__ h0,          // [2][32][8]
    float* __restrict__ ow)                // [32][1024][32][16]
{
    __shared__ float sW[384];
    __shared__ float sb[48];
    int tid = threadIdx.x;
    for (int i = tid; i < 384; i += 256) sW[i] = Whh[i];
    if (tid < 48) sb[tid] = bhh[tid];
    __syncthreads();

    int gtid = blockIdx.x * 256 + tid;     // 0..65535
    int b   = gtid & 31;
    int g   = (gtid >> 5) & 1023;          // (f*16+hk)*32 + l
    int dir = gtid >> 15;

    int f = g >> 9;
    int hk = (g >> 5) & 15;
    int l = g & 31;
    const int* tok = idx + ((((size_t)(b * 2 + f) * 16 + hk) * 32 + l) * 32);

    const float* W  = sW + dir * 192;
    const float* bb = sb + dir * 24;

    float h[8];
    #pragma unroll
    for (int k = 0; k < 8; ++k) h[k] = h0[(dir * 32 + b) * 8 + k];

    for (int p = 0; p < 32; ++p) {
        int t = dir ? (31 - p) : p;
        int token = tok[t];
        const float* gi = embproj + (size_t)token * 48 + dir * 24;

        float gr[8], gz[8], gn[8];
        #pragma unroll
        for (int j = 0; j < 8; ++j) {
            float ar = bb[j], az = bb[8 + j], an = bb[16 + j];
            #pragma unroll
            for (int k = 0; k < 8; ++k) {
                float hv = h[k];
                ar += hv * W[j * 8 + k];
                az += hv * W[(8 + j) * 8 + k];
                an += hv * W[(16 + j) * 8 + k];
            }
            gr[j] = ar; gz[j] = az; gn[j] = an;
        }
        #pragma unroll
        for (int j = 0; j < 8; ++j) {
            float r = sigmoidf_(gi[j] + gr[j]);
            float z = sigmoidf_(gi[8 + j] + gz[j]);
            float n = tanhf(gi[16 + j] + r * gn[j]);
            h[j] = (1.f - z) * n + z * h[j];
        }
        float* o = ow + (((size_t)(t * 1024 + g) * 32 + b) * 16 + dir * 8);
        *(float4*)o       = make_float4(h[0], h[1], h[2], h[3]);
        *(float4*)(o + 4) = make_float4(h[4], h[5], h[6], h[7]);
    }
}

// ---------------------------------------------------------------------------
// Generic bidirectional GRU with precomputed gi. 2*G*32 threads.
// gi_all rows: t*(G*32) + grp*32 + b ; out: [t][grp][b][16]
// ---------------------------------------------------------------------------
__global__ void gru_direct(
    const float* __restrict__ gi_all, const float* __restrict__ Whh,
    const float* __restrict__ bhh, const float* __restrict__ h0,
    float* __restrict__ out, int T, int G)
{
    __shared__ float sW[384];
    __shared__ float sb[48];
    int tid = threadIdx.x;
    for (int i = tid; i < 384; i += blockDim.x) sW[i] = Whh[i];
    if (tid < 48) sb[tid] = bhh[tid];
    __syncthreads();

    int gtid = blockIdx.x * blockDim.x + tid;
    int nseq = G * 32;
    if (gtid >= 2 * nseq) return;
    int b   = gtid & 31;
    int grp = (gtid >> 5) % G;
    int dir = (gtid >> 5) / G;

    const float* W  = sW + dir * 192;
    const float* bb = sb + dir * 24;

    float h[8];
    #pragma unroll
    for (int k = 0; k < 8; ++k) h[k] = h0[(dir * 32 + b) * 8 + k];

    for (int p = 0; p < T; ++p) {
        int t = dir ? (T - 1 - p) : p;
        const float* gi = gi_all + ((size_t)t * nseq + grp * 32 + b) * 48 + dir * 24;

        float gr[8], gz[8], gn[8];
        #pragma unroll
        for (int j = 0; j < 8; ++j) {
            float ar = bb[j], az = bb[8 + j], an = bb[16 + j];
            #pragma unroll
            for (int k = 0; k < 8; ++k) {
                float hv = h[k];
                ar += hv * W[j * 8 + k];
                az += hv * W[(8 + j) * 8 + k];
                an += hv * W[(16 + j) * 8 + k];
            }
            gr[j] = ar; gz[j] = az; gn[j] = an;
        }
        #pragma unroll
        for (int j = 0; j < 8; ++j) {
            float r = sigmoidf_(gi[j] + gr[j]);
            float z = sigmoidf_(gi[8 + j] + gz[j]);
            float n = tanhf(gi[16 + j] + r * gn[j]);
            h[j] = (1.f - z) * n + z * h[j];
        }
        float* o = out + (((size_t)(t * G + grp) * 32 + b) * 16 + dir * 8);
        *(float4*)o       = make_float4(h[0], h[1], h[2], h[3]);
        *(float4*)(o + 4) = make_float4(h[4], h[5], h[6], h[7]);
    }
}

// ---------------------------------------------------------------------------
// Fused attention score + softmax(over b) + weighted sum over t.
// Block = one group g, 512 threads: wave w handles d=w (16 waves), lane = b.
// Output row remapped into the NEXT level's GEMM layout:
//   row = (g % divv)*mulA + (g / divv)*mulB + b
// ---------------------------------------------------------------------------
__global__ __launch_bounds__(512) void attn_pool(
    const float* __restrict__ inp,   // [T][G][32][16]
    const float* __restrict__ Mc, const float* __restrict__ cc,
    float* __restrict__ outp, int T, int G, int divv, int mulA, int mulB)
{
    __shared__ float tile[512];
    __shared__ float sM[256];
    __shared__ float sc[16];
    int tid = threadIdx.x;
    int g = blockIdx.x;
    if (tid < 256) sM[tid] = Mc[tid];
    if (tid < 16)  sc[tid] = cc[tid];

    int d = tid >> 5;     // wave id = output feature
    int b = tid & 31;     // lane = batch
    float acc = 0.f;

    for (int t = 0; t < T; ++t) {
        __syncthreads();
        tile[tid] = inp[(size_t)(t * G + g) * 512 + tid];
        __syncthreads();
        float s = sc[d];
        #pragma unroll
        for (int k = 0; k < 16; ++k) s += tile[b * 16 + k] * sM[d * 16 + k];
        // softmax over b within the wave (wave32)
        float m = s;
        #pragma unroll
        for (int off = 16; off; off >>= 1) m = fmaxf(m, __shfl_xor(m, off));
        float e = __expf(s - m);
        float sum = e;
        #pragma unroll
        for (int off = 16; off; off >>= 1) sum += __shfl_xor(sum, off);
        acc += (e / sum) * tile[b * 16 + d];
    }
    int row = (g % divv) * mulA + (g / divv) * mulB + b;
    outp[(size_t)row * 16 + d] = acc;
}

// out[b][f*16+d] = files_a[f][b][d] - files_r[f][b][d]
__global__ __launch_bounds__(256) void final_sub(
    const float* __restrict__ fa, const float* __restrict__ fr, float* __restrict__ out)
{
    int tid = blockIdx.x * 256 + threadIdx.x;   // 0..1023
    int d = tid & 15;
    int f = (tid >> 4) & 1;
    int b = tid >> 5;
    int src = ((f * 32 + b) << 4) + d;
    out[((b * 2 + f) << 4) + d] = fa[src] - fr[src];
}

// ---------------------------------------------------------------------------
extern "C" void kernel_launch(void* const* d_in, const int* in_sizes, int n_in,
                              void* d_out, int out_size, void* d_ws, size_t ws_size,
                              hipStream_t stream)
{
    (void)in_sizes; (void)n_in; (void)out_size; (void)ws_size;
    const int*   added    = (const int*)d_in[0];
    const int*   removed  = (const int*)d_in[1];
    const float* hid_hunk = (const float*)d_in[2];
    const float* hid_sent = (const float*)d_in[3];
    const float* hid_word = (const float*)d_in[4];
    const float* embW     = (const float*)d_in[5];
    const float* w_Wih = (const float*)d_in[6];
    const float* w_Whh = (const float*)d_in[7];
    const float* w_bih = (const float*)d_in[8];
    const float* w_bhh = (const float*)d_in[9];
    const float* w_aW  = (const float*)d_in[10];
    const float* w_ab  = (const float*)d_in[11];
    const float* w_cW  = (const float*)d_in[12];
    const float* s_Wih = (const float*)d_in[13];
    const float* s_Whh = (const float*)d_in[14];
    const float* s_bih = (const float*)d_in[15];
    const float* s_bhh = (const float*)d_in[16];
    const float* s_aW  = (const float*)d_in[17];
    const float* s_ab  = (const float*)d_in[18];
    const float* s_cW  = (const float*)d_in[19];
    const float* h_Wih = (const float*)d_in[20];
    const float* h_Whh = (const float*)d_in[21];
    const float* h_bih = (const float*)d_in[22];
    const float* h_bhh = (const float*)d_in[23];
    const float* h_aW  = (const float*)d_in[24];
    const float* h_ab  = (const float*)d_in[25];
    const float* h_cW  = (const float*)d_in[26];
    float* out = (float*)d_out;

    // workspace carve-up (~104 MB of floats)
    float* ws = (float*)d_ws;
    size_t off = 0;
    float* embproj = ws + off; off += (size_t)VPAD * 48;         // 26.1 MB (padded)
    float* McW = ws + off; off += 256;  float* ccW = ws + off; off += 16;
    float* McS = ws + off; off += 256;  float* ccS = ws + off; off += 16;
    float* McH = ws + off; off += 256;  float* ccH = ws + off; off += 16;
    float* ow  = ws + off; off += (size_t)32 * 1024 * 32 * 16;   // 64 MB (L2-resident)
    float* xs  = ws + off; off += (size_t)32768 * 16;
    float* gis = ws + off; off += (size_t)32768 * 48;
    float* osb = ws + off; off += (size_t)32 * 32 * 32 * 16;
    float* xh  = ws + off; off += (size_t)1024 * 16;
    float* gih = ws + off; off += (size_t)1024 * 48;
    float* oh  = ws + off; off += (size_t)16 * 2 * 32 * 16;
    float* fa  = ws + off; off += 1024;
    float* fr  = ws + off; off += 1024;

    // fused attention matrices for the three levels
    attn_prep<<<1, 256, 0, stream>>>(w_aW, w_ab, w_cW, McW, ccW);
    attn_prep<<<1, 256, 0, stream>>>(s_aW, s_ab, s_cW, McS, ccS);
    attn_prep<<<1, 256, 0, stream>>>(h_aW, h_ab, h_cW, McH, ccH);

    // shared vocab-level input projection (serves both code tensors):
    // embproj[v][48] = embW[v] @ [Wih_f|Wih_b]^T + bias   (WMMA)
    {
        int mtiles = (VOCAB + 15) / 16;          // 8511
        int waves  = mtiles * 3;                 // 25533
        int blocks = (waves + 7) / 8;            // 8 waves / 256-thread block
        gemm_gi<<<blocks, 256, 0, stream>>>(embW, VOCAB, w_Wih, w_bih, embproj);
    }

    for (int c = 0; c < 2; ++c) {
        const int* code = c ? removed : added;
        float* files = c ? fr : fa;

        // word level: 65536 recurrent chains, T=32
        gru_word<<<256, 256, 0, stream>>>(code, embproj, w_Whh, w_bhh, hid_word, ow);
        // pool -> xs rows already in sentence-GEMM layout (l*1024 + fh*32 + b)
        attn_pool<<<1024, 512, 0, stream>>>(ow, McW, ccW, xs, 32, 1024, 32, 1024, 32);

        // sentence level
        gemm_gi<<<768, 256, 0, stream>>>(xs, 32768, s_Wih, s_bih, gis);   // 2048 mtiles * 3
        gru_direct<<<8, 256, 0, stream>>>(gis, s_Whh, s_bhh, hid_sent, osb, 32, 32);
        attn_pool<<<32, 512, 0, stream>>>(osb, McS, ccS, xh, 32, 32, 16, 64, 32);

        // hunk level
        gemm_gi<<<24, 256, 0, stream>>>(xh, 1024, h_Wih, h_bih, gih);     // 64 mtiles * 3
        gru_direct<<<1, 128, 0, stream>>>(gih, h_Whh, h_bhh, hid_hunk, oh, 16, 2);
        attn_pool<<<2, 512, 0, stream>>>(oh, McH, ccH, files, 16, 2, 16, 32, 0);
    }

    final_sub<<<4, 256, 0, stream>>>(fa, fr, out);
}